// CudaTensorProduct_17635135717499
// MI455X (gfx1250) — compile-verified
//
#include <hip/hip_runtime.h>
#include <hip/hip_bf16.h>
#include <stdint.h>

#define D1 288
#define D2 9
#define BT 16          // batch tile per workgroup
#define STRIDE 20      // LDS row stride in floats: multiple of 4 (16B align), not 16 (banks)
#define NSPLIT 2       // output-range splits (grid.y)

// ---------------------------------------------------------------------------
// Setup: pack {i1,i2,val} into int4 meta[], build CSR rowptr[] over sorted out_idx
// ---------------------------------------------------------------------------
__global__ void tp_setup_kernel(const int* __restrict__ i1, const int* __restrict__ i2,
                                const float* __restrict__ vals, const int* __restrict__ outi,
                                int nnz, int outd, int* __restrict__ rowptr,
                                int4* __restrict__ meta) {
    int t = blockIdx.x * 256 + threadIdx.x;
    if (meta != nullptr && t < nnz) {
        int4 m;
        m.x = i1[t];
        m.y = i2[t];
        m.z = __float_as_int(vals[t]);
        m.w = 0;
        meta[t] = m;
    }
    if (rowptr != nullptr && t <= outd) {
        // rowptr[t] = first k with outi[k] >= t  (outi is sorted non-decreasing)
        int lo = 0, hi = nnz;
        while (lo < hi) {
            int mid = (lo + hi) >> 1;
            if (outi[mid] < t) lo = mid + 1; else hi = mid;
        }
        rowptr[t] = lo;
    }
}

__device__ __forceinline__ int lower_bound_dev(const int* __restrict__ a, int n, int key) {
    int lo = 0, hi = n;
    while (lo < hi) {
        int mid = (lo + hi) >> 1;
        if (a[mid] < key) lo = mid + 1; else hi = mid;
    }
    return lo;
}

// ---------------------------------------------------------------------------
// Main: one workgroup = 16 batch rows staged to LDS (transposed) via
// gfx1250 async-to-LDS loads; lane owns output column o, streams its nnz
// segment once for all 16 rows (metadata amortized 16x).
// ---------------------------------------------------------------------------
template<bool USE_META, bool USE_ROWPTR>
__global__ __launch_bounds__(256)
void tp_main_kernel(const float* __restrict__ in1, const float* __restrict__ in2,
                    const int* __restrict__ rowptr, const int4* __restrict__ meta,
                    const int* __restrict__ i1x, const int* __restrict__ i2x,
                    const float* __restrict__ vals, const int* __restrict__ outi,
                    int nnz, float* __restrict__ out, int B, int outd, int ospan) {
    __shared__ __align__(16) float s1[D1 * STRIDE];   // [i][bt], stride 20 floats
    __shared__ __align__(16) float s2[D2 * STRIDE];

    const int tid = threadIdx.x;
    const int b0  = blockIdx.x * BT;

    // ---- stage in1 tile (transposed) via async-tensor path: global -> LDS ----
    for (int idx = tid; idx < BT * D1; idx += 256) {
        int bt = idx / D1;
        int i  = idx - bt * D1;                       // consecutive tid -> consecutive i: coalesced
        if (b0 + bt < B) {
            unsigned lds = (unsigned)(uintptr_t)(&s1[i * STRIDE + bt]);
            const float* g = in1 + (size_t)(b0 + bt) * D1 + i;
            asm volatile("global_load_async_to_lds_b32 %0, %1, off"
                         :: "v"(lds), "v"(g) : "memory");
        }
    }
    // ---- stage in2 tile (transposed) ----
    for (int idx = tid; idx < BT * D2; idx += 256) {
        int bt = idx / D2;
        int i  = idx - bt * D2;
        if (b0 + bt < B) {
            unsigned lds = (unsigned)(uintptr_t)(&s2[i * STRIDE + bt]);
            const float* g = in2 + (size_t)(b0 + bt) * D2 + i;
            asm volatile("global_load_async_to_lds_b32 %0, %1, off"
                         :: "v"(lds), "v"(g) : "memory");
        }
    }
    asm volatile("s_wait_asynccnt 0x0" ::: "memory");
    __syncthreads();

    const bool fullTile = (b0 + BT <= B);
    const int oBase = blockIdx.y * ospan;
    const int oEnd  = min(oBase + ospan, outd);

    for (int o = oBase + tid; o < oEnd; o += 256) {
        int kBeg, kEnd;
        if (USE_ROWPTR) {
            kBeg = rowptr[o];
            kEnd = rowptr[o + 1];
        } else {
            kBeg = lower_bound_dev(outi, nnz, o);
            kEnd = lower_bound_dev(outi, nnz, o + 1);
        }

        float acc[BT];
        #pragma unroll
        for (int bt = 0; bt < BT; ++bt) acc[bt] = 0.0f;

        for (int k = kBeg; k < kEnd; ++k) {
            int mi1, mi2;
            float v;
            if (USE_META) {
                int4 m = meta[k];                      // single b128, L2-resident
                mi1 = m.x; mi2 = m.y; v = __int_as_float(m.z);
            } else {
                mi1 = i1x[k]; mi2 = i2x[k]; v = vals[k];
            }
            const float4* a  = (const float4*)(&s1[mi1 * STRIDE]);  // 80B stride: 16B aligned
            const float4* b2 = (const float4*)(&s2[mi2 * STRIDE]);
            #pragma unroll
            for (int q = 0; q < 4; ++q) {
                float4 av = a[q];                      // ds_load_b128
                float4 bv = b2[q];                     // ds_load_b128
                acc[q * 4 + 0] = fmaf(v * av.x, bv.x, acc[q * 4 + 0]);
                acc[q * 4 + 1] = fmaf(v * av.y, bv.y, acc[q * 4 + 1]);
                acc[q * 4 + 2] = fmaf(v * av.z, bv.z, acc[q * 4 + 2]);
                acc[q * 4 + 3] = fmaf(v * av.w, bv.w, acc[q * 4 + 3]);
            }
        }

        // coalesced streaming stores: 32 consecutive o across the wave per row
        if (fullTile) {
            #pragma unroll
            for (int bt = 0; bt < BT; ++bt)
                __builtin_nontemporal_store(acc[bt], &out[(size_t)(b0 + bt) * outd + o]);
        } else {
            #pragma unroll
            for (int bt = 0; bt < BT; ++bt)
                if (b0 + bt < B)
                    __builtin_nontemporal_store(acc[bt], &out[(size_t)(b0 + bt) * outd + o]);
        }
    }
}

// ---------------------------------------------------------------------------
extern "C" void kernel_launch(void* const* d_in, const int* in_sizes, int n_in,
                              void* d_out, int out_size, void* d_ws, size_t ws_size,
                              hipStream_t stream) {
    if (n_in < 6) return;
    const float* in1   = (const float*)d_in[0];
    const float* in2   = (const float*)d_in[1];
    const float* vals  = (const float*)d_in[2];
    const int*   i1    = (const int*)d_in[3];
    const int*   i2    = (const int*)d_in[4];
    const int*   outi  = (const int*)d_in[5];

    const int B    = in_sizes[0] / D1;       // 8192
    const int nnz  = in_sizes[3];
    const int outd = out_size / B;           // CB_HEIGHT (2496)

    // workspace layout: [rowptr: (outd+1) ints][256B-aligned][meta: nnz int4]
    const size_t rowptrBytes = (size_t)(outd + 1) * sizeof(int);
    const size_t metaOff     = (rowptrBytes + 255) & ~(size_t)255;
    const size_t needMeta    = metaOff + (size_t)nnz * sizeof(int4);

    const bool haveRowptr = (d_ws != nullptr) && (ws_size >= rowptrBytes);
    const bool haveMeta   = (d_ws != nullptr) && (ws_size >= needMeta);

    int*  rowptr = haveRowptr ? (int*)d_ws : nullptr;
    int4* meta   = haveMeta   ? (int4*)((char*)d_ws + metaOff) : nullptr;

    if (haveRowptr || haveMeta) {
        int work = (nnz > outd + 1) ? nnz : (outd + 1);
        int blocks = (work + 255) / 256;
        tp_setup_kernel<<<blocks, 256, 0, stream>>>(i1, i2, vals, outi, nnz, outd, rowptr, meta);
    }

    int tiles = (B + BT - 1) / BT;                       // 512
    int ospan = (outd + NSPLIT - 1) / NSPLIT;            // 1248
    dim3 grid(tiles, NSPLIT);

    if (haveMeta) {
        tp_main_kernel<true, true><<<grid, 256, 0, stream>>>(
            in1, in2, rowptr, meta, i1, i2, vals, outi, nnz, (float*)d_out, B, outd, ospan);
    } else if (haveRowptr) {
        tp_main_kernel<false, true><<<grid, 256, 0, stream>>>(
            in1, in2, rowptr, meta, i1, i2, vals, outi, nnz, (float*)d_out, B, outd, ospan);
    } else {
        tp_main_kernel<false, false><<<grid, 256, 0, stream>>>(
            in1, in2, rowptr, meta, i1, i2, vals, outi, nnz, (float*)d_out, B, outd, ospan);
    }
}